// PointCloudShotDescriptor_9775345566000
// MI455X (gfx1250) — compile-verified
//
#include <hip/hip_runtime.h>

typedef __attribute__((ext_vector_type(2))) float v2f;
typedef __attribute__((ext_vector_type(8))) float v8f;

#define NPTS  4096
#define NCLD  8
#define KNN   16
#define NBINS 352
#define EPSF  1e-8f
#define PI_F  3.14159265358979323846f
#define WAVES 4

// insert (d2v, civ) into unsorted top-16 (bd/bi) kept in registers,
// tracking current max value/position.  Fully unrolled -> v_cndmask chains.
#define TOPK_INSERT(d2v, civ)                                        \
  do { if ((d2v) < curmax) {                                         \
    _Pragma("unroll")                                                \
    for (int _j = 0; _j < 16; ++_j)                                  \
      if (_j == curpos) { bd[_j] = (d2v); bi[_j] = (civ); }          \
    curmax = bd[0]; curpos = 0;                                      \
    _Pragma("unroll")                                                \
    for (int _j = 1; _j < 16; ++_j)                                  \
      if (bd[_j] > curmax) { curmax = bd[_j]; curpos = _j; }         \
  } } while (0)

__global__ __launch_bounds__(32) void shot_init_kernel(unsigned int* rad_bits) {
  if (threadIdx.x < NCLD) rad_bits[threadIdx.x] = 0u;  // 0.0f bits
}

// One 4-wave block per 16-query tile.  Whole cloud (48KB) staged in LDS via
// async-to-LDS copies; Gram tiles via WMMA f32 16x16x4; per-wave candidate
// ranges merged through LDS.
__global__ __launch_bounds__(128) void knn_lrf_kernel(
    const float* __restrict__ verts,     // (8,4096,3)
    float* __restrict__ normals,         // (8,4096,3)
    int* __restrict__ knn_idx,           // (8,4096,16)
    unsigned int* __restrict__ rad_bits) // (8,)
{
  __shared__ float sv[NPTS * 3];          // staged cloud: 49152 B
  __shared__ float ldsG[WAVES * 256];     // per-wave 16x16 Gram tile
  __shared__ float mrgD[16 * 48];         // merge buffer (waves 1..3)
  __shared__ int   mrgI[16 * 48];

  const int tid   = threadIdx.x;
  const int wave  = tid >> 5;
  const int lane  = tid & 31;
  const int row   = lane & 15;
  const bool lo   = lane < 16;
  const int colbase = lo ? 0 : 8;

  const int cloud = blockIdx.x >> 8;
  const int qtile = blockIdx.x & 255;
  const float* V  = verts + (size_t)cloud * NPTS * 3;

  // ---- stage whole cloud into LDS with async copies (ASYNCcnt) ----
  {
    const unsigned long long gbase = (unsigned long long)(uintptr_t)V;
    const unsigned sv_base = (unsigned)(uintptr_t)sv;  // low 32b of flat = LDS offset
    for (int k = 0; k < 24; ++k) {                     // 24*128*16B = 49152 B
      const int goff = (k * 128 + tid) * 16;
      const unsigned ldsoff = sv_base + (unsigned)goff;
      asm volatile("global_load_async_to_lds_b128 %0, %1, %2"
                   :: "v"(ldsoff), "v"(goff), "s"(gbase) : "memory");
    }
    asm volatile("s_wait_asynccnt 0x0" ::: "memory");
  }
  __syncthreads();

  // query point of this lane's row
  const int q = qtile * 16 + row;
  const float qx = sv[q*3+0], qy = sv[q*3+1], qz = sv[q*3+2];
  const float rq = qx*qx + qy*qy + qz*qz;

  // A matrix (16x4 f32): lanes 0-15 hold K0,K1 ; lanes 16-31 hold K2,K3
  v2f a;
  a.x = lo ? qx : qz;
  a.y = lo ? qy : 0.0f;

  float bd[16]; int bi[16];
#pragma unroll
  for (int j = 0; j < 16; ++j) { bd[j] = 3.4e38f; bi[j] = 0; }
  float curmax = 3.4e38f; int curpos = 0;

  float* gG = &ldsG[wave * 256];
  const int t0 = wave * 64, t1 = t0 + 64;   // disjoint candidate range per wave

  for (int t = t0; t < t1; ++t) {
    const int cand = t * 16 + row;
    const float cx = sv[cand*3+0], cy = sv[cand*3+1], cz = sv[cand*3+2];
    const float rc = cx*cx + cy*cy + cz*cz;

    // B matrix (4x16 f32): VGPR0 = {K0 | K1}, VGPR1 = {K2 | K3} across halves
    v2f b;
    b.x = lo ? cx : cy;
    b.y = lo ? cz : 0.0f;

    v8f c = {};
    c = __builtin_amdgcn_wmma_f32_16x16x4_f32(
        /*neg_a=*/false, a, /*neg_b=*/false, b,
        /*c_mod=*/(short)0, c, /*reuse_a=*/false, /*reuse_b=*/false);

    // D layout: VGPR v -> M = v (lanes 0-15) or v+8 (lanes 16-31), N = lane&15
#pragma unroll
    for (int v = 0; v < 8; ++v) {
      const int m = v + (lo ? 0 : 8);
      gG[m*16 + row] = c[v];
    }

    // selection: lane owns query row (lane&15); halves split the 16 columns
#pragma unroll
    for (int cc = 0; cc < 8; ++cc) {
      const int col = colbase + cc;
      const float g   = gG[row*16 + col];
      const float rcc = __shfl(rc, col, 32);
      const float d2  = rq + rcc - 2.0f * g;
      const int   ci  = t*16 + col;
      TOPK_INSERT(d2, ci);
    }
  }

  // intra-wave merge: hi-half list into lo-half list
#pragma unroll
  for (int j = 0; j < 16; ++j) {
    const float d  = __shfl(bd[j], row + 16, 32);
    const int   ci = __shfl(bi[j], row + 16, 32);
    if (lo) TOPK_INSERT(d, ci);
  }

  // inter-wave merge through LDS: waves 1..3 publish, wave 0 consumes
  if (wave != 0 && lo) {
    const int base = row * 48 + (wave - 1) * 16;
#pragma unroll
    for (int j = 0; j < 16; ++j) { mrgD[base + j] = bd[j]; mrgI[base + j] = bi[j]; }
  }
  __syncthreads();

  if (wave == 0 && lo) {
    const int base = row * 48;
#pragma unroll 1
    for (int e = 0; e < 48; ++e) {
      const float d  = mrgD[base + e];
      const int   ci = mrgI[base + e];
      TOPK_INSERT(d, ci);
    }

    const float radius = sqrtf(fmaxf(curmax, 0.0f));

    // weighted covariance of relative neighbor coords (gathered from LDS)
    float sw = 0.f, c00=0.f,c01=0.f,c02=0.f,c11=0.f,c12=0.f,c22=0.f;
#pragma unroll
    for (int j = 0; j < 16; ++j) {
      const int ni = bi[j];
      const float nx = sv[ni*3+0] - qx;
      const float ny = sv[ni*3+1] - qy;
      const float nz = sv[ni*3+2] - qz;
      const float w  = fmaxf(radius - sqrtf(fmaxf(bd[j], 0.0f)), 0.0f);
      sw  += w;
      c00 += w*nx*nx; c01 += w*nx*ny; c02 += w*nx*nz;
      c11 += w*ny*ny; c12 += w*ny*nz; c22 += w*nz*nz;
    }
    const float inv = 1.0f / (sw + EPSF);
    c00*=inv; c01*=inv; c02*=inv; c11*=inv; c12*=inv; c22*=inv;

    // smallest eigenvalue of symmetric 3x3 (trigonometric closed form)
    const float qtr = (c00 + c11 + c22) * (1.0f/3.0f);
    const float p1  = c01*c01 + c02*c02 + c12*c12;
    const float b00 = c00 - qtr, b11 = c11 - qtr, b22 = c22 - qtr;
    const float p2  = b00*b00 + b11*b11 + b22*b22 + 2.0f*p1;
    const float p   = sqrtf(p2 * (1.0f/6.0f)) + 1e-20f;
    const float ip  = 1.0f / p;
    const float B00=b00*ip, B01=c01*ip, B02=c02*ip;
    const float B11=b11*ip, B12=c12*ip, B22=b22*ip;
    const float detB = B00*(B11*B22 - B12*B12)
                     - B01*(B01*B22 - B12*B02)
                     + B02*(B01*B12 - B11*B02);
    const float rdet = fminf(fmaxf(0.5f*detB, -1.0f), 1.0f);
    const float phi  = acosf(rdet) * (1.0f/3.0f);
    const float lmin = qtr + 2.0f*p*cosf(phi + 2.0943951023931953f); // +2pi/3

    // eigenvector: largest cross product of rows of (C - lmin*I)
    const float m00 = c00 - lmin, m11 = c11 - lmin, m22 = c22 - lmin;
    float e0x = c01*c12 - c02*m11, e0y = c02*c01 - m00*c12, e0z = m00*m11 - c01*c01;
    float e1x = c01*m22 - c02*c12, e1y = c02*c02 - m00*m22, e1z = m00*c12 - c01*c02;
    float e2x = m11*m22 - c12*c12, e2y = c12*c02 - c01*m22, e2z = c01*c12 - m11*c02;
    const float n0 = e0x*e0x + e0y*e0y + e0z*e0z;
    const float n1 = e1x*e1x + e1y*e1y + e1z*e1z;
    const float n2 = e2x*e2x + e2y*e2y + e2z*e2z;
    float ex, ey, ez, nn;
    if (n0 >= n1 && n0 >= n2)      { ex=e0x; ey=e0y; ez=e0z; nn=n0; }
    else if (n1 >= n2)             { ex=e1x; ey=e1y; ez=e1z; nn=n1; }
    else                           { ex=e2x; ey=e2y; ez=e2z; nn=n2; }
    if (nn < 1e-30f) { ex = 1.0f; ey = 0.0f; ez = 0.0f; nn = 1.0f; }
    const float rn = rsqrtf(nn);
    ex *= rn; ey *= rn; ez *= rn;

    // sign disambiguation: sum of signs of projections
    float s = 0.0f;
#pragma unroll
    for (int j = 0; j < 16; ++j) {
      const int ni = bi[j];
      const float nx = sv[ni*3+0] - qx;
      const float ny = sv[ni*3+1] - qy;
      const float nz = sv[ni*3+2] - qz;
      const float d  = nx*ex + ny*ey + nz*ez;
      s += (d > 0.0f ? 1.0f : 0.0f) - (d < 0.0f ? 1.0f : 0.0f);
    }
    if (s < 0.0f) { ex = -ex; ey = -ey; ez = -ez; }

    const size_t gp = (size_t)cloud * NPTS + q;
    normals[gp*3+0] = ex; normals[gp*3+1] = ey; normals[gp*3+2] = ez;
#pragma unroll
    for (int j = 0; j < 16; ++j) knn_idx[gp*KNN + j] = bi[j];

    atomicMax(&rad_bits[cloud], __float_as_uint(radius)); // valid: radius >= 0
  }
}

// One thread per point: 352-bin SHOT histogram, L2-normalized.
__global__ __launch_bounds__(256) void shot_descr_kernel(
    const float* __restrict__ verts,
    const float* __restrict__ normals,
    const int* __restrict__ knn_idx,
    const unsigned int* __restrict__ rad_bits,
    float* __restrict__ out)
{
  const int gid = blockIdx.x * 256 + threadIdx.x;   // 0..32767
  const int cloud = gid >> 12;
  const int p     = gid & (NPTS - 1);

  const float* V   = verts   + (size_t)cloud * NPTS * 3;
  const float* Nrm = normals + (size_t)cloud * NPTS * 3;
  const float radius = __uint_as_float(rad_bits[cloud]);

  const float px = V[p*3+0], py = V[p*3+1], pz = V[p*3+2];
  const float ax = Nrm[p*3+0], ay = Nrm[p*3+1], az0 = Nrm[p*3+2];

  int   flat[KNN];
  int   msk[KNN];
#pragma unroll
  for (int j = 0; j < KNN; ++j) {
    const int ni = knn_idx[((size_t)cloud * NPTS + p) * KNN + j];
    const float rx = V[ni*3+0] - px;
    const float ry = V[ni*3+1] - py;
    const float rz = V[ni*3+2] - pz;
    const float bx = Nrm[ni*3+0], by = Nrm[ni*3+1], bz = Nrm[ni*3+2];
    const float cs = fminf(fmaxf(ax*bx + ay*by + az0*bz, -1.0f), 1.0f);
    const float r  = sqrtf(rx*rx + ry*ry + rz*rz);

    const float azim = atan2f(ry, rx);
    int azb = (int)((azim + PI_F) * (8.0f / (2.0f * PI_F)));
    azb = min(max(azb, 0), 7);
    const float el = (rz / (r + EPSF) + 1.0f) * 0.5f;
    int elb = (int)(el * 2.0f);                  elb = min(max(elb, 0), 1);
    int rb  = (int)(r / (radius + EPSF) * 2.0f); rb  = min(max(rb, 0), 1);
    int hb  = (int)((cs + 1.0f) * 0.5f * 11.0f); hb  = min(max(hb, 0), 10);

    flat[j] = ((azb * 2 + elb) * 2 + rb) * 11 + hb;
    msk[j]  = (r <= radius) ? 1 : 0;
  }

  // per-neighbor bin counts; norm^2 = sum over bins of count^2
  int   cnt[KNN];
  float n2 = 0.0f;
#pragma unroll
  for (int i = 0; i < KNN; ++i) {
    int ci = 0;
#pragma unroll
    for (int j = 0; j < KNN; ++j)
      ci += (flat[j] == flat[i]) ? msk[j] : 0;
    cnt[i] = ci;
    n2 += msk[i] ? (float)ci : 0.0f;
  }
  const float invn = 1.0f / (sqrtf(n2) + EPSF);

  float* orow = out + (size_t)gid * NBINS;
  float4 z4; z4.x = z4.y = z4.z = z4.w = 0.0f;
#pragma unroll 4
  for (int b = 0; b < NBINS / 4; ++b) ((float4*)orow)[b] = z4;

#pragma unroll
  for (int i = 0; i < KNN; ++i) {
    bool first = (msk[i] != 0);
#pragma unroll
    for (int j = 0; j < KNN; ++j)
      if (j < i && msk[j] && flat[j] == flat[i]) first = false;
    if (first) orow[flat[i]] = (float)cnt[i] * invn;
  }
}

extern "C" void kernel_launch(void* const* d_in, const int* in_sizes, int n_in,
                              void* d_out, int out_size, void* d_ws, size_t ws_size,
                              hipStream_t stream) {
  (void)in_sizes; (void)n_in; (void)out_size; (void)ws_size;
  const float* verts = (const float*)d_in[0];
  float* out = (float*)d_out;

  char* ws = (char*)d_ws;
  float* normals      = (float*)ws;                                  // 8*4096*3 f32
  int*   knn_idx      = (int*)(ws + (size_t)NCLD*NPTS*3*sizeof(float));
  unsigned int* radb  = (unsigned int*)(ws + (size_t)NCLD*NPTS*3*sizeof(float)
                                           + (size_t)NCLD*NPTS*KNN*sizeof(int));

  shot_init_kernel<<<1, 32, 0, stream>>>(radb);
  knn_lrf_kernel<<<NCLD * (NPTS / 16), 128, 0, stream>>>(verts, normals, knn_idx, radb);
  shot_descr_kernel<<<(NCLD * NPTS) / 256, 256, 0, stream>>>(verts, normals, knn_idx, radb, out);
}